// Model_20418274525430
// MI455X (gfx1250) — compile-verified
//
#include <hip/hip_runtime.h>
#include <hip/hip_bf16.h>
#include <math.h>

// ---------------- problem constants (from setup_inputs) ----------------
#define T_DIM 512
#define N_NODES 1024
#define H_DIM 256
#define Z_DIM 128
#define D_DIM 256         // all pooled projections land in H_DIM=256
#define SAMPLE_NUM 10
#define TIMESPAN 4
#define NEG_DIST 85       // 512 // 6
#define NCE_START 64      // 512 // 8
#define NCE_NC 351        // end - start = 415 - 64
#define EPSV 1e-8f

typedef __attribute__((ext_vector_type(2))) float v2f;
typedef __attribute__((ext_vector_type(4))) float v4f;
typedef __attribute__((ext_vector_type(8))) float v8f;

// ---------------- wave32 reduction helper ----------------
__device__ __forceinline__ float wred(float v) {
    v += __shfl_xor(v, 16, 32);
    v += __shfl_xor(v, 8, 32);
    v += __shfl_xor(v, 4, 32);
    v += __shfl_xor(v, 2, 32);
    v += __shfl_xor(v, 1, 32);
    return v;
}

// ---------------- streaming mean over nodes: all_h [T,1024,256] -> [T,256]
// One block per t; 1024 threads = 16 node-groups x 64 float4 columns.
// Per-lane b128 non-temporal loads: 512B/wave/VMEM-op, unroll-8 keeps 8
// loads (128B/lane) in flight. 537MB streamed once -> NT keeps L2 clean.
__global__ __launch_bounds__(1024) void mean_h_kernel(
    const float* __restrict__ all_h, float* __restrict__ h_mean) {
    __shared__ v4f part[16][64];
    const int t = blockIdx.x;
    const int q = threadIdx.x & 63;        // float4 column (64 * 4 = 256 feats)
    const int p = threadIdx.x >> 6;        // 0..15 node group, 64 nodes each
    const v4f* base = (const v4f*)(all_h + (size_t)t * N_NODES * H_DIM)
                      + (size_t)(p * 64) * 64 + q;   // row stride = 64 v4f
    v4f s = {0.f, 0.f, 0.f, 0.f};
#pragma unroll 8
    for (int n = 0; n < 64; ++n)
        s += __builtin_nontemporal_load(base + (size_t)n * 64);
    part[p][q] = s;
    __syncthreads();
    if (threadIdx.x < 64) {
        v4f tot = part[0][threadIdx.x];
#pragma unroll
        for (int r = 1; r < 16; ++r) tot += part[r][threadIdx.x];
        tot = tot * (1.0f / N_NODES);
        *((v4f*)(h_mean + (size_t)t * H_DIM) + threadIdx.x) = tot;
    }
}

// all_z [T,1024,128] -> [T,128]; 512 threads = 16 groups x 32 float4 columns.
__global__ __launch_bounds__(512) void mean_z_kernel(
    const float* __restrict__ all_z, float* __restrict__ z_mean) {
    __shared__ v4f part[16][32];
    const int t = blockIdx.x;
    const int q = threadIdx.x & 31;        // float4 column (32 * 4 = 128 feats)
    const int p = threadIdx.x >> 5;        // 0..15 node group, 64 nodes each
    const v4f* base = (const v4f*)(all_z + (size_t)t * N_NODES * Z_DIM)
                      + (size_t)(p * 64) * 32 + q;   // row stride = 32 v4f
    v4f s = {0.f, 0.f, 0.f, 0.f};
#pragma unroll 8
    for (int n = 0; n < 64; ++n)
        s += __builtin_nontemporal_load(base + (size_t)n * 32);
    part[p][q] = s;
    __syncthreads();
    if (threadIdx.x < 32) {
        v4f tot = part[0][threadIdx.x];
#pragma unroll
        for (int r = 1; r < 16; ++r) tot += part[r][threadIdx.x];
        tot = tot * (1.0f / N_NODES);
        *((v4f*)(z_mean + (size_t)t * Z_DIM) + threadIdx.x) = tot;
    }
}

// ---------------- fp32 WMMA GEMM:  Out[M,256] = A[M,K] @ W[256,K]^T + bias
// One wave (32 threads) per 16x16 output tile; K-loop of V_WMMA_F32_16X16X4_F32.
// ISA lane layout: A 16x4 -> lane half selects K pair {half*2, half*2+1}, row = lane%16;
// B 4x16 -> same K pair, col = lane%16; C/D VGPR v -> row v + 8*half, col = lane%16.
__global__ __launch_bounds__(32) void wmma_gemm_awt(
    const float* __restrict__ A, const float* __restrict__ W,
    const float* __restrict__ bias, float* __restrict__ Out, int K) {
    const int N = D_DIM;
    const int lane = threadIdx.x;
    const int half = lane >> 4;
    const int l16  = lane & 15;
    const int m = blockIdx.x * 16 + l16;   // A row feeding this lane
    const int n = blockIdx.y * 16 + l16;   // W row (= output col) feeding this lane
    const float* arow = A + (size_t)m * K;
    const float* wrow = W + (size_t)n * K;
    v8f acc = {};
#pragma unroll 8
    for (int k0 = 0; k0 < K; k0 += 4) {
        const int ka = k0 + half * 2;      // even -> 8B aligned v2f loads
        v2f a = *(const v2f*)(arow + ka);
        v2f b = *(const v2f*)(wrow + ka);
        acc = __builtin_amdgcn_wmma_f32_16x16x4_f32(
            false, a, false, b, (short)0, acc, false, false);
    }
    const float bv = bias[n];
#pragma unroll
    for (int v = 0; v < 8; ++v) {
        const int om = blockIdx.x * 16 + v + half * 8;
        Out[(size_t)om * N + n] = acc[v] + bv;
    }
}

// ---------------- per-row L2 norms of z_pool [512,256] -> z_norm [512]
__global__ __launch_bounds__(32) void znorm_kernel(
    const float* __restrict__ z_pool, float* __restrict__ z_norm) {
    const int t = blockIdx.x;
    const int lane = threadIdx.x;
    const float* row = z_pool + (size_t)t * D_DIM;
    float s = 0.f;
#pragma unroll
    for (int j = 0; j < 8; ++j) { float v = row[lane + 32 * j]; s += v * v; }
    s = wred(s);
    if (lane == 0) z_norm[t] = sqrtf(s);
}

// ---------------- distance = sum_t ||z_t - mean||^2 / T = sum(z^2)/T - ||mean||^2
__global__ __launch_bounds__(256) void dist_kernel(
    const float* __restrict__ z_pool, float* __restrict__ out) {
    __shared__ float red[256];
    const int d = threadIdx.x;
    float s = 0.f, s2 = 0.f;
    for (int t = 0; t < T_DIM; ++t) {
        float v = z_pool[(size_t)t * D_DIM + d];
        s += v; s2 += v * v;
    }
    const float gm = s * (1.0f / T_DIM);
    red[d] = s2 * (1.0f / T_DIM) - gm * gm;
    __syncthreads();
    for (int w = 128; w > 0; w >>= 1) {
        if (d < w) red[d] += red[d + w];
        __syncthreads();
    }
    if (d == 0) out[1] = red[0];
}

// ---------------- NCE: one block per candidate t0, one wave per timespan i.
// cos(samples, c_phi) over D=256, log-softmax pick-0; per-(nc,s) logp0 to ws.
__global__ __launch_bounds__(128) void nce_kernel(
    const float* __restrict__ z_pool, const float* __restrict__ z_norm,
    const float* __restrict__ c_all, float* __restrict__ logp0) {
    const int nc = blockIdx.x;
    const int t0 = NCE_START + nc;
    const int wave = threadIdx.x >> 5;     // 0..3 -> i = wave+1
    const int lane = threadIdx.x & 31;
    const float* crow = c_all + (size_t)t0 * D_DIM;
    float c[8]; float cn2 = 0.f;
#pragma unroll
    for (int j = 0; j < 8; ++j) { c[j] = crow[lane + 32 * j]; cn2 += c[j] * c[j]; }
    cn2 = wred(cn2);
    const float cnorm = fmaxf(sqrtf(cn2), EPSV);
    const int i = wave + 1;
    float cosv[SAMPLE_NUM];
#pragma unroll
    for (int k = 0; k < SAMPLE_NUM; ++k) {
        const int idx = (k == 0) ? (t0 + i) : (t0 + i + NEG_DIST + (k - 1));
        const float* zrow = z_pool + (size_t)idx * D_DIM;
        float dsum = 0.f;
#pragma unroll
        for (int j = 0; j < 8; ++j) dsum += zrow[lane + 32 * j] * c[j];
        dsum = wred(dsum);
        const float sn = fmaxf(z_norm[idx], EPSV);
        cosv[k] = dsum / (sn * cnorm);
    }
    if (lane == 0) {
        float mx = cosv[0];
#pragma unroll
        for (int k = 1; k < SAMPLE_NUM; ++k) mx = fmaxf(mx, cosv[k]);
        float se = 0.f;
#pragma unroll
        for (int k = 0; k < SAMPLE_NUM; ++k) se += expf(cosv[k] - mx);
        logp0[nc * TIMESPAN + wave] = cosv[0] - mx - logf(se);
    }
}

// ---------------- final deterministic reduction of logp0 -> nce_loss
__global__ __launch_bounds__(256) void nce_reduce_kernel(
    const float* __restrict__ logp0, float* __restrict__ out) {
    __shared__ float red[256];
    const int tid = threadIdx.x;
    float s = 0.f;
    for (int idx = tid; idx < NCE_NC * TIMESPAN; idx += 256) s += logp0[idx];
    red[tid] = s;
    __syncthreads();
    for (int w = 128; w > 0; w >>= 1) {
        if (tid < w) red[tid] += red[tid + w];
        __syncthreads();
    }
    if (tid == 0) out[0] = red[0] * (-1.0f / (float)(NCE_NC * TIMESPAN));
}

extern "C" void kernel_launch(void* const* d_in, const int* in_sizes, int n_in,
                              void* d_out, int out_size, void* d_ws, size_t ws_size,
                              hipStream_t stream) {
    const float* all_h = (const float*)d_in[0];
    const float* all_z = (const float*)d_in[1];
    const float* Wh    = (const float*)d_in[2];
    const float* bh    = (const float*)d_in[3];
    const float* Wz    = (const float*)d_in[4];
    const float* bz    = (const float*)d_in[5];
    const float* Wc    = (const float*)d_in[6];
    const float* bc    = (const float*)d_in[7];
    float* out = (float*)d_out;

    // workspace layout (floats)
    float* ws     = (float*)d_ws;
    float* h_mean = ws;                              // 512*256
    float* z_mean = h_mean + T_DIM * H_DIM;          // 512*128
    float* h_pool = z_mean + T_DIM * Z_DIM;          // 512*256
    float* z_pool = h_pool + T_DIM * D_DIM;          // 512*256
    float* c_all  = z_pool + T_DIM * D_DIM;          // 512*256
    float* z_norm = c_all  + T_DIM * D_DIM;          // 512
    float* logp0  = z_norm + T_DIM;                  // 351*4

    // 1) bandwidth-bound node pooling (the ~805MB stream; everything else is noise)
    mean_h_kernel<<<T_DIM, 1024, 0, stream>>>(all_h, h_mean);
    mean_z_kernel<<<T_DIM, 512, 0, stream>>>(all_z, z_mean);

    // 2) fp32 WMMA projections (16x16 tile per wave; grid = M/16 x N/16)
    dim3 gtiles(T_DIM / 16, D_DIM / 16);
    wmma_gemm_awt<<<gtiles, 32, 0, stream>>>(h_mean, Wh, bh, h_pool, H_DIM);
    wmma_gemm_awt<<<gtiles, 32, 0, stream>>>(z_mean, Wz, bz, z_pool, Z_DIM);
    wmma_gemm_awt<<<gtiles, 32, 0, stream>>>(h_pool, Wc, bc, c_all, H_DIM);

    // 3) distance (writes out[1]) and sample-row norms
    znorm_kernel<<<T_DIM, 32, 0, stream>>>(z_pool, z_norm);
    dist_kernel<<<1, 256, 0, stream>>>(z_pool, out);

    // 4) NCE cosine / log-softmax, then deterministic reduce (writes out[0])
    nce_kernel<<<NCE_NC, 128, 0, stream>>>(z_pool, z_norm, c_all, logp0);
    nce_reduce_kernel<<<1, 256, 0, stream>>>(logp0, out);
}